// GATNetwork_86174223827365
// MI455X (gfx1250) — compile-verified
//
#include <hip/hip_runtime.h>
#include <hip/hip_bf16.h>
#include <math.h>

typedef float v2f __attribute__((ext_vector_type(2)));
typedef float v8f __attribute__((ext_vector_type(8)));

// ---------------------------------------------------------------------------
// Dense linear: Y[n, cout] = X[n, cin] @ W[cout, cin]^T + B
// One wave per 16x16 output tile, V_WMMA_F32_16X16X4_F32, K stepped by 4.
// Fragment layouts per CDNA5 ISA 7.12.2 (32-bit A 16x4 / B 4x16 / C,D 16x16).
// ---------------------------------------------------------------------------
__global__ void k_linear_wmma(const float* __restrict__ X, const float* __restrict__ W,
                              const float* __restrict__ B, float* __restrict__ Y,
                              int nrows, int cin, int cout) {
    int gid  = blockIdx.x * blockDim.x + threadIdx.x;
    int wave = gid >> 5;
    int lane = gid & 31;
    int colTiles = cout >> 4;
    int rowTiles = nrows >> 4;
    if (wave >= colTiles * rowTiles) return;   // wave-uniform exit
    int row0 = (wave / colTiles) << 4;
    int col0 = (wave % colTiles) << 4;
    int half = lane >> 4;      // selects K pair {0,1} vs {2,3}
    int m    = lane & 15;      // M index for A, N index for B

    const float* xrow = X + (size_t)(row0 + m) * cin + 2 * half;
    const float* wrow = W + (size_t)(col0 + m) * cin + 2 * half;

    v8f c = {};
    for (int k = 0; k < cin; k += 4) {
        v2f a = *(const v2f*)(xrow + k);   // A[m][k+2*half], A[m][k+2*half+1]
        v2f b = *(const v2f*)(wrow + k);   // B^T: W[n][k+2*half], W[n][k+2*half+1]
        c = __builtin_amdgcn_wmma_f32_16x16x4_f32(false, a, false, b,
                                                  (short)0, c, false, false);
    }
    float bv = B[col0 + m];
    float* yout = Y + (size_t)(row0 + 8 * half) * cout + col0 + m;
#pragma unroll
    for (int r = 0; r < 8; ++r)
        yout[(size_t)r * cout] = c[r] + bv;   // D row = row0 + r + 8*half, col = col0 + m
}

// ---------------------------------------------------------------------------
// Per-layer scratch init: segment-max = -inf, denom = 0, accumulator = 0
// ---------------------------------------------------------------------------
__global__ void k_init_layer(float* __restrict__ maxb, float* __restrict__ den,
                             float* __restrict__ acc, int n, int h, int hc) {
    int i = blockIdx.x * blockDim.x + threadIdx.x;
    if (i < n * h) { maxb[i] = -INFINITY; den[i] = 0.0f; }
    if (i < n * hc) acc[i] = 0.0f;
}

// ---------------------------------------------------------------------------
// Edge pass A: logits[e][h] = att[h] . leaky_relu(xl[src] + xr[dst]); segment max.
// One wave per edge, lane = channel (C == 32). Self-loops appended at e >= E.
// ---------------------------------------------------------------------------
__global__ void k_edge_logits(const float* __restrict__ xl, const float* __restrict__ xr,
                              const int* __restrict__ ei, const float* __restrict__ att,
                              float* __restrict__ logits, float* __restrict__ maxb,
                              int E, int N, int H) {
    int gid  = blockIdx.x * blockDim.x + threadIdx.x;
    int e    = gid >> 5;
    int lane = gid & 31;
    int E2   = E + N;
    if (e >= E2) return;
    int s, d;
    if (e < E) { s = ei[e]; d = ei[E + e]; } else { s = e - E; d = s; }
    s = min(max(s, 0), N - 1);
    d = min(max(d, 0), N - 1);
    int HC = H * 32;
    const float* pl = xl + (size_t)s * HC;
    const float* pr = xr + (size_t)d * HC;
    for (int h = 0; h < H; ++h) {
        float v = pl[h * 32 + lane] + pr[h * 32 + lane];
        v = (v > 0.0f) ? v : 0.2f * v;           // leaky_relu(0.2)
        v *= att[h * 32 + lane];
#pragma unroll
        for (int off = 16; off > 0; off >>= 1) v += __shfl_xor(v, off, 32);
        if (lane == 0) {
            logits[(size_t)e * H + h] = v;
            atomicMax(maxb + (size_t)d * H + h, v);
        }
    }
}

// ---------------------------------------------------------------------------
// Edge pass B: ex = exp(logit - max[dst]); denom[dst] += ex;
//              acc[dst] += ex * xl[src]   (normalized later by denom+1e-16)
// ---------------------------------------------------------------------------
__global__ void k_edge_agg(const float* __restrict__ xl, const int* __restrict__ ei,
                           const float* __restrict__ logits, const float* __restrict__ maxb,
                           float* __restrict__ den, float* __restrict__ acc,
                           int E, int N, int H) {
    int gid  = blockIdx.x * blockDim.x + threadIdx.x;
    int e    = gid >> 5;
    int lane = gid & 31;
    int E2   = E + N;
    if (e >= E2) return;
    int s, d;
    if (e < E) { s = ei[e]; d = ei[E + e]; } else { s = e - E; d = s; }
    s = min(max(s, 0), N - 1);
    d = min(max(d, 0), N - 1);
    int HC = H * 32;
    const float* pl = xl + (size_t)s * HC;
    float* pa = acc + (size_t)d * HC;
    for (int h = 0; h < H; ++h) {
        float ex = __expf(logits[(size_t)e * H + h] - maxb[(size_t)d * H + h]);
        if (lane == 0) atomicAdd(den + (size_t)d * H + h, ex);
        atomicAdd(pa + h * 32 + lane, ex * pl[h * 32 + lane]);
    }
}

// ---------------------------------------------------------------------------
// Node finalize (concat heads): out = acc/(den+1e-16) + bias, optional relu (in place)
// ---------------------------------------------------------------------------
__global__ void k_finalize_concat(const float* __restrict__ acc, const float* __restrict__ den,
                                  const float* __restrict__ bias, float* __restrict__ out,
                                  int N, int H, int relu) {
    int HC = H * 32;
    int i = blockIdx.x * blockDim.x + threadIdx.x;
    if (i >= N * HC) return;
    int n = i / HC, j = i - n * HC, h = j >> 5;
    float v = acc[i] / (den[(size_t)n * H + h] + 1e-16f) + bias[j];
    out[i] = relu ? fmaxf(v, 0.0f) : v;
}

// Node finalize (mean over heads, layer 3): out[n][c] = mean_h(acc/(den+eps)) + bias[c]
__global__ void k_finalize_mean(const float* __restrict__ acc, const float* __restrict__ den,
                                const float* __restrict__ bias, float* __restrict__ out,
                                int N, int H) {
    int i = blockIdx.x * blockDim.x + threadIdx.x;
    if (i >= N * 32) return;
    int n = i >> 5, c = i & 31;
    float s = 0.0f;
    for (int h = 0; h < H; ++h)
        s += acc[(size_t)n * H * 32 + h * 32 + c] / (den[(size_t)n * H + h] + 1e-16f);
    out[i] = s / (float)H + bias[c];
}

// ---------------------------------------------------------------------------
// Column-wise pooling pass 1: sum, sum^2, min, max (+ max of t*x for softmax aggr)
// Block of 256 = D columns x (256/D) row groups; coalesced rows, tree-reduce, atomics.
// ---------------------------------------------------------------------------
__device__ __forceinline__ float blk_reduce(float v, float* sbuf, int tid, int D, int op) {
    sbuf[tid] = v; __syncthreads();
    for (int s = 128; s >= D; s >>= 1) {
        if (tid < s) {
            float a = sbuf[tid], b = sbuf[tid + s];
            sbuf[tid] = (op == 0) ? (a + b) : (op == 1) ? fminf(a, b) : fmaxf(a, b);
        }
        __syncthreads();
    }
    float r = sbuf[tid]; __syncthreads();
    return r;
}

__global__ void k_pool1(const float* __restrict__ X, int N, int D,
                        const float* __restrict__ tptr, int useT,
                        float* __restrict__ sum, float* __restrict__ sumsq,
                        float* __restrict__ mn, float* __restrict__ mx,
                        float* __restrict__ mxs) {
    __shared__ float sbuf[256];
    int tid = threadIdx.x;
    int col = tid % D, rg = tid / D, rpb = 256 / D;
    float tval = useT ? tptr[0] : 1.0f;
    float lsum = 0.f, lsq = 0.f, lmn = INFINITY, lmx = -INFINITY, lms = -INFINITY;
    for (int r = blockIdx.x * rpb + rg; r < N; r += gridDim.x * rpb) {
        float v = X[(size_t)r * D + col];
        lsum += v; lsq += v * v;
        lmn = fminf(lmn, v); lmx = fmaxf(lmx, v);
        if (useT) lms = fmaxf(lms, tval * v);
    }
    float t0 = blk_reduce(lsum, sbuf, tid, D, 0);
    if (tid < D) atomicAdd(sum + col, t0);
    float t1 = blk_reduce(lsq, sbuf, tid, D, 0);
    if (tid < D) atomicAdd(sumsq + col, t1);
    float t2 = blk_reduce(lmn, sbuf, tid, D, 1);
    if (tid < D) atomicMin(mn + col, t2);
    float t3 = blk_reduce(lmx, sbuf, tid, D, 2);
    if (tid < D) atomicMax(mx + col, t3);
    if (useT) {
        float t4 = blk_reduce(lms, sbuf, tid, D, 2);
        if (tid < D) atomicMax(mxs + col, t4);
    }
}

// Pooling pass 2 (softmax aggregation): sexp += exp(t*x - mxs); swx += exp(..)*x
__global__ void k_pool2(const float* __restrict__ X, int N, int D,
                        const float* __restrict__ tptr, const float* __restrict__ mxs,
                        float* __restrict__ sexp, float* __restrict__ swx) {
    __shared__ float sbuf[256];
    int tid = threadIdx.x;
    int col = tid % D, rg = tid / D, rpb = 256 / D;
    float tval = tptr[0];
    float m = mxs[col];
    float lse = 0.f, lwx = 0.f;
    for (int r = blockIdx.x * rpb + rg; r < N; r += gridDim.x * rpb) {
        float v = X[(size_t)r * D + col];
        float ex = __expf(tval * v - m);
        lse += ex; lwx += ex * v;
    }
    float t0 = blk_reduce(lse, sbuf, tid, D, 0);
    if (tid < D) atomicAdd(sexp + col, t0);
    float t1 = blk_reduce(lwx, sbuf, tid, D, 0);
    if (tid < D) atomicAdd(swx + col, t1);
}

// Stats buffer layout (floats):
//   [0..63] xsum [64..127] xsumsq [128..191] xmin [192..255] xmax
//   [256..287] hsum [288..319] hsumsq [320..351] hmin [352..383] hmax
//   [384..415] hmaxs [416..447] hsexp [448..479] hswx
__global__ void k_init_stats(float* __restrict__ stats) {
    int t = threadIdx.x;  // 512 threads
    float v = 0.0f;
    if ((t >= 128 && t < 192) || (t >= 320 && t < 352)) v = INFINITY;          // mins
    else if ((t >= 192 && t < 256) || (t >= 352 && t < 416)) v = -INFINITY;    // maxes
    stats[t] = v;
}

// Build z[512] = concat(g[192], globfeat[320]) with nan_to_num(1,1,1)
__global__ void k_build_z(const float* __restrict__ stats, int N, float* __restrict__ z) {
    int t = threadIdx.x;
    const float* xsum = stats;        const float* xsq = stats + 64;
    const float* xmn  = stats + 128;  const float* xmx = stats + 192;
    const float* hsum = stats + 256;  const float* hsq = stats + 288;
    const float* hmn  = stats + 320;  const float* hmx = stats + 352;
    const float* hse  = stats + 416;  const float* hwx = stats + 448;
    float fN = (float)N, v = 0.0f;
    if (t < 192) {
        int c = t & 31, part = t >> 5;
        if (part == 0) v = hsum[c];
        else if (part == 1) v = hsum[c] / fN;
        else if (part == 2) { float m = hsum[c] / fN, m2 = hsq[c] / fN;
                              v = sqrtf(fmaxf(m2 - m * m, 0.0f) + 1e-5f); }
        else if (part == 3) v = hmn[c];
        else if (part == 4) v = hmx[c];
        else                v = hwx[c] / hse[c];
    } else {
        int u = t - 192; int c = u % 64, part = u / 64;
        if (part == 0) v = xsum[c];
        else if (part == 1) v = xsum[c] / fN;
        else if (part == 2) { float m = xsum[c] / fN, m2 = xsq[c] / fN;
                              v = sqrtf(fmaxf(m2 - m * m, 0.0f) + 1e-5f); }
        else if (part == 3) v = xmn[c];
        else                v = xmx[c];
    }
    if (isnan(v) || isinf(v)) v = 1.0f;   // nan_to_num(nan=1, posinf=1, neginf=1)
    z[t] = v;
}

// Fused MLP: z[512] -> relu(fc1 512) -> relu(fcm 256) -> fc2 16. One block, 512 thr.
__global__ void k_mlp(const float* __restrict__ z,
                      const float* __restrict__ w1, const float* __restrict__ b1,
                      const float* __restrict__ wm, const float* __restrict__ bm,
                      const float* __restrict__ w2, const float* __restrict__ b2,
                      float* __restrict__ out) {
    __shared__ float zb[512], h1[512], h2[256];
    int t = threadIdx.x;
    zb[t] = z[t]; __syncthreads();
    {
        const float* wr = w1 + (size_t)t * 512;
        float a = b1[t];
        for (int k = 0; k < 512; ++k) a += wr[k] * zb[k];
        h1[t] = fmaxf(a, 0.0f);
    }
    __syncthreads();
    if (t < 256) {
        const float* wr = wm + (size_t)t * 512;
        float a = bm[t];
        for (int k = 0; k < 512; ++k) a += wr[k] * h1[k];
        h2[t] = fmaxf(a, 0.0f);
    }
    __syncthreads();
    if (t < 16) {
        const float* wr = w2 + (size_t)t * 256;
        float a = b2[t];
        for (int k = 0; k < 256; ++k) a += wr[k] * h2[k];
        out[t] = a;
    }
}

// ---------------------------------------------------------------------------
extern "C" void kernel_launch(void* const* d_in, const int* in_sizes, int n_in,
                              void* d_out, int out_size, void* d_ws, size_t ws_size,
                              hipStream_t stream) {
    const float* x   = (const float*)d_in[0];
    const int*   ei  = (const int*)d_in[1];
    const float* W1l = (const float*)d_in[2];  const float* b1l = (const float*)d_in[3];
    const float* W1r = (const float*)d_in[4];  const float* b1r = (const float*)d_in[5];
    const float* at1 = (const float*)d_in[6];  const float* cb1 = (const float*)d_in[7];
    const float* W2l = (const float*)d_in[8];  const float* b2l = (const float*)d_in[9];
    const float* W2r = (const float*)d_in[10]; const float* b2r = (const float*)d_in[11];
    const float* at2 = (const float*)d_in[12]; const float* cb2 = (const float*)d_in[13];
    const float* W3l = (const float*)d_in[14]; const float* b3l = (const float*)d_in[15];
    const float* W3r = (const float*)d_in[16]; const float* b3r = (const float*)d_in[17];
    const float* at3 = (const float*)d_in[18]; const float* cb3 = (const float*)d_in[19];
    const float* tpt = (const float*)d_in[20];
    const float* f1w = (const float*)d_in[21]; const float* f1b = (const float*)d_in[22];
    const float* fmw = (const float*)d_in[23]; const float* fmb = (const float*)d_in[24];
    const float* f2w = (const float*)d_in[25]; const float* f2b = (const float*)d_in[26];

    const int N = in_sizes[0] / 64;
    const int E = in_sizes[1] / 2;
    const int E2 = E + N;

    float* ws = (float*)d_ws;
    size_t o = 0;
    float* xl    = ws + o; o += (size_t)N * 128;
    float* xr    = ws + o; o += (size_t)N * 128;
    float* h1    = ws + o; o += (size_t)N * 128;
    float* h2    = ws + o; o += (size_t)N * 128;
    float* h3    = ws + o; o += (size_t)N * 32;
    float* maxb  = ws + o; o += (size_t)N * 4;
    float* den   = ws + o; o += (size_t)N * 4;
    float* lg    = ws + o; o += (size_t)E2 * 4;
    float* stats = ws + o; o += 512;
    float* z     = ws + o; o += 512;

    const int TB = 256;
    const int eb = (E2 * 32 + TB - 1) / TB;   // one wave per edge

    // Input-feature pooling stats (globfeat) + h-pooling stats init
    k_init_stats<<<1, 512, 0, stream>>>(stats);
    k_pool1<<<256, TB, 0, stream>>>(x, N, 64, tpt, 0,
                                    stats + 0, stats + 64, stats + 128, stats + 192, nullptr);

    // ---- GAT layer 1: 64 -> 4x32 (concat, relu) ----
    {
        int H = 4, HC = 128, cin = 64;
        int tiles = (N / 16) * (HC / 16);
        int lb = (tiles * 32 + TB - 1) / TB;
        k_linear_wmma<<<lb, TB, 0, stream>>>(x, W1l, b1l, xl, N, cin, HC);
        k_linear_wmma<<<lb, TB, 0, stream>>>(x, W1r, b1r, xr, N, cin, HC);
        k_init_layer<<<(N * HC + TB - 1) / TB, TB, 0, stream>>>(maxb, den, h1, N, H, HC);
        k_edge_logits<<<eb, TB, 0, stream>>>(xl, xr, ei, at1, lg, maxb, E, N, H);
        k_edge_agg<<<eb, TB, 0, stream>>>(xl, ei, lg, maxb, den, h1, E, N, H);
        k_finalize_concat<<<(N * HC + TB - 1) / TB, TB, 0, stream>>>(h1, den, cb1, h1, N, H, 1);
    }
    // ---- GAT layer 2: 128 -> 4x32 (concat, relu) ----
    {
        int H = 4, HC = 128, cin = 128;
        int tiles = (N / 16) * (HC / 16);
        int lb = (tiles * 32 + TB - 1) / TB;
        k_linear_wmma<<<lb, TB, 0, stream>>>(h1, W2l, b2l, xl, N, cin, HC);
        k_linear_wmma<<<lb, TB, 0, stream>>>(h1, W2r, b2r, xr, N, cin, HC);
        k_init_layer<<<(N * HC + TB - 1) / TB, TB, 0, stream>>>(maxb, den, h2, N, H, HC);
        k_edge_logits<<<eb, TB, 0, stream>>>(xl, xr, ei, at2, lg, maxb, E, N, H);
        k_edge_agg<<<eb, TB, 0, stream>>>(xl, ei, lg, maxb, den, h2, E, N, H);
        k_finalize_concat<<<(N * HC + TB - 1) / TB, TB, 0, stream>>>(h2, den, cb2, h2, N, H, 1);
    }
    // ---- GAT layer 3: 128 -> 2x32 (mean over heads, no relu) ----
    {
        int H = 2, HC = 64, cin = 128;
        int tiles = (N / 16) * (HC / 16);
        int lb = (tiles * 32 + TB - 1) / TB;
        float* acc3 = h1;  // h1 is dead after layer 2
        k_linear_wmma<<<lb, TB, 0, stream>>>(h2, W3l, b3l, xl, N, cin, HC);
        k_linear_wmma<<<lb, TB, 0, stream>>>(h2, W3r, b3r, xr, N, cin, HC);
        k_init_layer<<<(N * HC + TB - 1) / TB, TB, 0, stream>>>(maxb, den, acc3, N, H, HC);
        k_edge_logits<<<eb, TB, 0, stream>>>(xl, xr, ei, at3, lg, maxb, E, N, H);
        k_edge_agg<<<eb, TB, 0, stream>>>(xl, ei, lg, maxb, den, acc3, E, N, H);
        k_finalize_mean<<<(N * 32 + TB - 1) / TB, TB, 0, stream>>>(acc3, den, cb3, h3, N, H);
    }

    // ---- h-pooling (sum/mean/std/min/max + softmax aggregation) ----
    k_pool1<<<256, TB, 0, stream>>>(h3, N, 32, tpt, 1,
                                    stats + 256, stats + 288, stats + 320, stats + 352, stats + 384);
    k_pool2<<<256, TB, 0, stream>>>(h3, N, 32, tpt, stats + 384, stats + 416, stats + 448);

    // ---- z = concat(g, globfeat); MLP head ----
    k_build_z<<<1, 512, 0, stream>>>(stats, N, z);
    k_mlp<<<1, 512, 0, stream>>>(z, f1w, f1b, fmw, fmb, f2w, f2b, (float*)d_out);
}